// BigramLMv2_65670049956185
// MI455X (gfx1250) — compile-verified
//
#include <hip/hip_runtime.h>
#include <math.h>

typedef float v2f __attribute__((ext_vector_type(2)));
typedef float v8f __attribute__((ext_vector_type(8)));

// D = A(16x4, f32) * B(4x16, f32) + C(16x16, f32), wave32.
// 8-arg pattern: (neg_a, A, neg_b, B, c_mod, C, reuse_a, reuse_b)
#define WMMA_F32(A, B, C) \
    __builtin_amdgcn_wmma_f32_16x16x4_f32(false, (A), false, (B), (short)0, (C), false, false)

#define T_CTX   8
#define D_EMBD  32
#define D_HEAD  8
#define VOCAB   65
#define WAVES   8
#define THREADS (WAVES * 32)
#define ROWS_PER_WAVE  16
#define ROWS_PER_BLOCK (WAVES * ROWS_PER_WAVE)   // 128
#define LG_STRIDE 66

__global__ __launch_bounds__(THREADS)
void bigram_attn_fused(const int* __restrict__ ix,
                       const int* __restrict__ targets,
                       const float* __restrict__ emb_W,
                       const float* __restrict__ pos_W,
                       const float* __restrict__ Wk,
                       const float* __restrict__ Wq,
                       const float* __restrict__ Wv,
                       const float* __restrict__ lm_W,
                       const float* __restrict__ lm_b,
                       float* __restrict__ out_logits,
                       float* __restrict__ block_loss)
{
    // ---- block-shared weights / tables (stay L2/LDS-hot; logits stream is NT) ----
    __shared__ float s_emb[VOCAB * D_EMBD];   // 8320 B
    __shared__ float s_pos[T_CTX * D_EMBD];   // 1024 B
    __shared__ float s_wqk[D_EMBD * 16];      // [k][n]: n<8 -> Wq, n>=8 -> Wk
    __shared__ float s_wv [D_EMBD * 16];      // [k][n]: n<8 -> Wv, else 0
    __shared__ float s_lmw[D_HEAD * VOCAB];   // row-major [k][col]
    __shared__ float s_lmb[80];               // bias padded to 80
    // ---- per-wave staging ----
    __shared__ float s_q[WAVES][16 * 8];      // Q, later reused for softmaxed wei
    __shared__ float s_k[WAVES][16 * 8];      // K, later reused for O
    __shared__ float s_v[WAVES][16 * 8];      // V
    __shared__ float s_lg[WAVES][16 * LG_STRIDE]; // logits for CE pass
    __shared__ float s_red[WAVES];

    const int tid = threadIdx.x;
    for (int i = tid; i < VOCAB * D_EMBD; i += THREADS) s_emb[i] = emb_W[i];
    for (int i = tid; i < T_CTX * D_EMBD; i += THREADS) s_pos[i] = pos_W[i];
    for (int i = tid; i < D_EMBD * 16; i += THREADS) {
        int k = i >> 4, n = i & 15;
        s_wqk[i] = (n < 8) ? Wq[k * 8 + n] : Wk[k * 8 + (n - 8)];
        s_wv[i]  = (n < 8) ? Wv[k * 8 + n] : 0.0f;
    }
    for (int i = tid; i < D_HEAD * VOCAB; i += THREADS) s_lmw[i] = lm_W[i];
    for (int i = tid; i < 80; i += THREADS) s_lmb[i] = (i < VOCAB) ? lm_b[i] : 0.0f;
    __syncthreads();

    const int wave = tid >> 5;
    const int lane = tid & 31;
    const int half = lane >> 4;       // 0: K-sub {0,1}, 1: K-sub {2,3} within a chunk
    const int l15  = lane & 15;
    const int waveRow = (blockIdx.x * WAVES + wave) * ROWS_PER_WAVE;  // global row base

    float* q = s_q[wave];
    float* kk = s_k[wave];
    float* v = s_v[wave];
    float* lg = s_lg[wave];

    // ================= Stage 1: X = emb[ix]+pos ; [Q|K] = X@Wqk ; V = X@Wv ====
    const int tok_id = __builtin_nontemporal_load(&ix[waveRow + l15]); // read-once stream
    const float* embRow = &s_emb[tok_id * D_EMBD];
    const float* posRow = &s_pos[(l15 & 7) * D_EMBD];

    v8f cqk = {0.f, 0.f, 0.f, 0.f, 0.f, 0.f, 0.f, 0.f};
    v8f cv  = {0.f, 0.f, 0.f, 0.f, 0.f, 0.f, 0.f, 0.f};
#pragma unroll
    for (int t = 0; t < 8; ++t) {                       // K = 32 -> 8 chunks of 4
        const int c0 = 4 * t + 2 * half;
        v2f a, bqk, bv;
        a.x = embRow[c0]     + posRow[c0];
        a.y = embRow[c0 + 1] + posRow[c0 + 1];
        bqk.x = s_wqk[c0 * 16 + l15];
        bqk.y = s_wqk[(c0 + 1) * 16 + l15];
        bv.x  = s_wv[c0 * 16 + l15];
        bv.y  = s_wv[(c0 + 1) * 16 + l15];
        cqk = WMMA_F32(a, bqk, cqk);
        cv  = WMMA_F32(a, bv,  cv);
    }
    // C layout: VGPR r, lane -> row = r + 8*half, col = l15
#pragma unroll
    for (int r = 0; r < 8; ++r) {
        const int row = r + 8 * half;
        if (l15 < 8) {
            q[row * 8 + l15] = cqk[r];
            v[row * 8 + l15] = cv[r];
        } else {
            kk[row * 8 + (l15 - 8)] = cqk[r];
        }
    }

    // ================= Stage 2: S = Q @ K^T (16x16, diag 8x8 blocks valid) ====
    v8f cs = {0.f, 0.f, 0.f, 0.f, 0.f, 0.f, 0.f, 0.f};
#pragma unroll
    for (int t = 0; t < 2; ++t) {                       // K = D_HEAD = 8
        const int c0 = 4 * t + 2 * half;
        v2f a, b;
        a.x = q[l15 * 8 + c0];     a.y = q[l15 * 8 + c0 + 1];   // A[m][k] = Q[m][k]
        b.x = kk[l15 * 8 + c0];    b.y = kk[l15 * 8 + c0 + 1];  // B[k][n] = K[n][k]
        cs = WMMA_F32(a, b, cs);
    }
    const float scale = 0.17677669529663687f;           // D_EMBD^-0.5 (as in reference)
#pragma unroll
    for (int r = 0; r < 8; ++r) {                       // keep only same-batch blocks
        const int row = r + 8 * half;
        if ((l15 >> 3) == half)
            q[row * 8 + (l15 & 7)] = cs[r] * scale;     // wei reuses s_q
    }

    // ================= Stage 3: causal softmax (one row per lane 0..15) =======
    if (lane < 16) {
        float* w = &q[lane * 8];
        const int tl = lane & 7;
        float tmp[8];
        float m = -INFINITY;
#pragma unroll
        for (int s = 0; s < 8; ++s) {
            tmp[s] = (s <= tl) ? w[s] : -INFINITY;
            m = fmaxf(m, tmp[s]);
        }
        float sum = 0.0f;
#pragma unroll
        for (int s = 0; s < 8; ++s) {
            const float e = __expf(tmp[s] - m);         // exp(-inf) = 0
            tmp[s] = e;
            sum += e;
        }
        const float inv = 1.0f / sum;
#pragma unroll
        for (int s = 0; s < 8; ++s) w[s] = tmp[s] * inv;
    }

    // ================= Stage 4: O = Wfull(16x16 block-diag) @ V(16x8) =========
    v8f co = {0.f, 0.f, 0.f, 0.f, 0.f, 0.f, 0.f, 0.f};
#pragma unroll
    for (int t = 0; t < 4; ++t) {                       // K = 16 (s over both batches)
        const int c0 = 4 * t + 2 * half;
        const bool same = ((c0 >> 3) == (l15 >> 3));    // c0,c0+1 share the >>3
        v2f a, b;
        a.x = same ? q[l15 * 8 + (c0 & 7)] : 0.0f;
        a.y = same ? q[l15 * 8 + ((c0 + 1) & 7)] : 0.0f;
        b.x = (l15 < 8) ? v[c0 * 8 + l15] : 0.0f;       // B[s][h] = V[s][h]
        b.y = (l15 < 8) ? v[(c0 + 1) * 8 + l15] : 0.0f;
        co = WMMA_F32(a, b, co);
    }
    float* o = kk;                                      // O reuses s_k
#pragma unroll
    for (int r = 0; r < 8; ++r) {
        const int row = r + 8 * half;
        if (l15 < 8) o[row * 8 + l15] = co[r];
    }

    // ================= Stage 5: logits = O @ lm_W + b, 5 N-tiles of 16 ========
#pragma unroll
    for (int n = 0; n < 5; ++n) {
        const int col = n * 16 + l15;
        const bool cok = (col < VOCAB);
        const float bias = cok ? s_lmb[col] : 0.0f;
        v8f c = {bias, bias, bias, bias, bias, bias, bias, bias};
#pragma unroll
        for (int t = 0; t < 2; ++t) {                   // K = D_HEAD = 8
            const int c0 = 4 * t + 2 * half;
            v2f a, b;
            a.x = o[l15 * 8 + c0];     a.y = o[l15 * 8 + c0 + 1];
            b.x = cok ? s_lmw[c0 * VOCAB + col] : 0.0f;
            b.y = cok ? s_lmw[(c0 + 1) * VOCAB + col] : 0.0f;
            c = WMMA_F32(a, b, c);
        }
#pragma unroll
        for (int r = 0; r < 8; ++r) {
            const int row = r + 8 * half;
            if (cok) {
                // 272 MB write-once stream > 192 MB L2: bypass with NT hint so the
                // shared emb/weight tables stay resident for all 8192 blocks.
                __builtin_nontemporal_store(
                    c[r], &out_logits[(size_t)(waveRow + row) * VOCAB + col]);
                lg[row * LG_STRIDE + col] = c[r];
            }
        }
    }

    // ================= Stage 6: cross-entropy partials =========================
    float rowloss = 0.0f;
    if (lane < 16) {
        const float* lr = &lg[lane * LG_STRIDE];
        float m = -INFINITY;
        for (int c2 = 0; c2 < VOCAB; ++c2) m = fmaxf(m, lr[c2]);
        float sum = 0.0f;
        for (int c2 = 0; c2 < VOCAB; ++c2) sum += __expf(lr[c2] - m);
        const int tgt = __builtin_nontemporal_load(&targets[waveRow + lane]);
        rowloss = (__logf(sum) + m) - lr[tgt];
    }
#pragma unroll
    for (int off = 16; off > 0; off >>= 1)
        rowloss += __shfl_down(rowloss, off, 32);
    if (lane == 0) s_red[wave] = rowloss;
    __syncthreads();
    if (tid == 0) {
        float bs = 0.0f;
        for (int w2 = 0; w2 < WAVES; ++w2) bs += s_red[w2];
        block_loss[blockIdx.x] = bs;
    }
}

__global__ __launch_bounds__(256)
void loss_reduce(const float* __restrict__ part, float* __restrict__ out_loss,
                 int nPart, float invN)
{
    __shared__ float red[256];
    float s = 0.0f;
    for (int i = threadIdx.x; i < nPart; i += 256) s += part[i];   // fixed order
    red[threadIdx.x] = s;
    __syncthreads();
    for (int off = 128; off > 0; off >>= 1) {
        if ((int)threadIdx.x < off) red[threadIdx.x] += red[threadIdx.x + off];
        __syncthreads();
    }
    if (threadIdx.x == 0) *out_loss = red[0] * invN;
}

extern "C" void kernel_launch(void* const* d_in, const int* in_sizes, int n_in,
                              void* d_out, int out_size, void* d_ws, size_t ws_size,
                              hipStream_t stream)
{
    // setup_inputs order: ix, targets, emb_W, pos_W, Wk, Wq, Wv, lm_W, lm_b
    const int*   ix      = (const int*)d_in[0];
    const int*   targets = (const int*)d_in[1];
    const float* emb_W   = (const float*)d_in[2];
    const float* pos_W   = (const float*)d_in[3];
    const float* Wk      = (const float*)d_in[4];
    const float* Wq      = (const float*)d_in[5];
    const float* Wv      = (const float*)d_in[6];
    const float* lm_W    = (const float*)d_in[7];
    const float* lm_b    = (const float*)d_in[8];

    const int rows   = in_sizes[0];              // B*T = 1,048,576
    const int blocks = rows / ROWS_PER_BLOCK;    // 8192

    float* out = (float*)d_out;
    float* ws  = (float*)d_ws;                   // needs blocks*4 B = 32 KB

    bigram_attn_fused<<<dim3(blocks), dim3(THREADS), 0, stream>>>(
        ix, targets, emb_W, pos_W, Wk, Wq, Wv, lm_W, lm_b, out, ws);

    loss_reduce<<<dim3(1), dim3(256), 0, stream>>>(
        ws, out + (size_t)rows * VOCAB, blocks, 1.0f / (float)rows);
}